// SwitchDropTokenCalculator_57114475102484
// MI455X (gfx1250) — compile-verified
//
#include <hip/hip_runtime.h>
#include <hip/hip_bf16.h>

#define NE 16
#define HD 4096
#define FF 688
#define NT 8192
#define CAP 640           // int(1.25 * 8192 / 16)

typedef float v2f __attribute__((ext_vector_type(2)));
typedef float v8f __attribute__((ext_vector_type(8)));

// ---------------------------------------------------------------------------
// Kernel 0: zero the output (dropped tokens must produce zeros; harness
// poisons d_out before timing).
// ---------------------------------------------------------------------------
__global__ void moe_zero_kernel(float4* __restrict__ p, size_t n4) {
    size_t i = (size_t)blockIdx.x * blockDim.x + threadIdx.x;
    size_t stride = (size_t)gridDim.x * blockDim.x;
    float4 z; z.x = 0.f; z.y = 0.f; z.z = 0.f; z.w = 0.f;
    for (; i < n4; i += stride) p[i] = z;
}

// ---------------------------------------------------------------------------
// Kernel 1: deterministic dispatch. One block per expert scans tokens in
// order, computing each token's rank within its expert (cumsum semantics of
// the reference). Tokens with rank < CAP are kept; rows[e*CAP+pos] = token.
// ---------------------------------------------------------------------------
__global__ __launch_bounds__(256) void moe_dispatch_kernel(
        const int* __restrict__ topk_idx, int* __restrict__ rows,
        int* __restrict__ cnt) {
    __shared__ int wsum[8];
    const int e = blockIdx.x;
    const int tid = threadIdx.x;
    const int w = tid >> 5;
    const int lane = tid & 31;
    int running = 0;
    for (int base = 0; base < NT; base += 256) {
        const int t = base + tid;
        const bool match = (topk_idx[t] == e);
        const unsigned mask = (unsigned)__ballot(match);   // wave32: low 32 bits
        const int lpre = __popc(mask & ((1u << lane) - 1u));
        if (lane == 0) wsum[w] = __popc(mask);
        __syncthreads();
        int woff = 0, tot = 0;
        for (int i = 0; i < 8; i++) {
            int c = wsum[i];
            if (i < w) woff += c;
            tot += c;
        }
        if (match) {
            const int pos = running + woff + lpre;
            if (pos < CAP) rows[e * CAP + pos] = t;
        }
        running += tot;
        __syncthreads();
    }
    if (tid == 0) cnt[e] = (running < CAP) ? running : CAP;
}

// ---------------------------------------------------------------------------
// Kernel 2: fused gate/up GEMM + SiLU.  Block = 256 threads = 8 waves.
// Block tile: M=64 (4 x 16 sub-tiles per wave), N=128 (one 16-col sub-tile
// per wave).  K marched in chunks of 32 (8 WMMA k-steps).  A staged in LDS
// (stride 34 pad -> conflict-free b64 fragment reads); B fragments loaded
// directly from global (L2-resident per expert).
// ---------------------------------------------------------------------------
__global__ __launch_bounds__(256) void moe_gemm1_kernel(
        const float* __restrict__ x,
        const float* __restrict__ Wg, const float* __restrict__ Wu,
        const int* __restrict__ rows, const int* __restrict__ cnt,
        float* __restrict__ H) {
    __shared__ __align__(16) float lds_a[64 * 34];
    __shared__ int tok[64];

    const int e  = blockIdx.z;
    const int ce = cnt[e];
    const int m0 = blockIdx.x * 64;
    if (m0 >= ce) return;

    const int tid  = threadIdx.x;
    const int w    = tid >> 5;
    const int lane = tid & 31;
    const int n0w  = blockIdx.y * 128 + w * 16;
    const bool wactive = (n0w < FF);

    if (tid < 64) {
        const int m = m0 + tid;
        tok[tid] = (m < ce) ? rows[e * CAP + m] : -1;
    }
    __syncthreads();

    const size_t wbase = (size_t)e * HD * FF;
    v8f accg[4] = {};
    v8f accu[4] = {};

    const int nc    = n0w + (lane & 15);
    const int khalf = (lane >> 4) * 2;

    for (int k0 = 0; k0 < HD; k0 += 32) {
        __syncthreads();
        // cooperative A load: wave loads one full 32-wide row per iteration
        #pragma unroll
        for (int i = 0; i < 8; i++) {
            const int m = w + i * 8;
            const int t = tok[m];
            float v = 0.f;
            if (t >= 0) v = x[(size_t)t * HD + k0 + lane];
            lds_a[m * 34 + lane] = v;
        }
        __syncthreads();

        if (wactive) {
            #pragma unroll
            for (int ks = 0; ks < 32; ks += 4) {
                const int kk = k0 + ks + khalf;
                const float* bgp = Wg + wbase + (size_t)kk * FF + nc;
                const float* bup = Wu + wbase + (size_t)kk * FF + nc;
                v2f bg; bg.x = bgp[0]; bg.y = bgp[FF];
                v2f bu; bu.x = bup[0]; bu.y = bup[FF];
                #pragma unroll
                for (int ms = 0; ms < 4; ms++) {
                    const float* ap =
                        &lds_a[(ms * 16 + (lane & 15)) * 34 + ks + khalf];
                    v2f af; af.x = ap[0]; af.y = ap[1];
                    accg[ms] = __builtin_amdgcn_wmma_f32_16x16x4_f32(
                        false, af, false, bg, (short)0, accg[ms], false, false);
                    accu[ms] = __builtin_amdgcn_wmma_f32_16x16x4_f32(
                        false, af, false, bu, (short)0, accu[ms], false, false);
                }
            }
        }
    }

    if (wactive) {
        float* hb = H + (size_t)e * CAP * FF;
        const int colbase = n0w + (lane & 15);
        const int rowoff  = (lane >> 4) * 8;
        #pragma unroll
        for (int ms = 0; ms < 4; ms++) {
            #pragma unroll
            for (int j = 0; j < 8; j++) {
                const int row = m0 + ms * 16 + rowoff + j;
                if (row < ce) {
                    const float g = accg[ms][j];
                    const float u = accu[ms][j];
                    const float h = (g / (1.f + __expf(-g))) * u;  // silu(g)*u
                    hb[(size_t)row * FF + colbase] = h;
                }
            }
        }
    }
}

// ---------------------------------------------------------------------------
// Kernel 3: down GEMM (K = 688 = 43 exact chunks of 16) + scatter with
// gate-score scaling. Same tiling as GEMM1; A (H rows) staged with stride-18
// pad. Output scattered to d_out[token] (zero kernel already covered drops).
// ---------------------------------------------------------------------------
__global__ __launch_bounds__(256) void moe_gemm2_kernel(
        const float* __restrict__ H, const float* __restrict__ Wd,
        const int* __restrict__ rows, const int* __restrict__ cnt,
        const float* __restrict__ scores, float* __restrict__ out) {
    __shared__ __align__(16) float lds_a[64 * 18];
    __shared__ int tok[64];

    const int e  = blockIdx.z;
    const int ce = cnt[e];
    const int m0 = blockIdx.x * 64;
    if (m0 >= ce) return;

    const int tid  = threadIdx.x;
    const int w    = tid >> 5;
    const int lane = tid & 31;
    const int n0w  = blockIdx.y * 128 + w * 16;   // D=4096 = 32*128, always valid

    if (tid < 64) {
        const int m = m0 + tid;
        tok[tid] = (m < ce) ? rows[e * CAP + m] : -1;
    }
    __syncthreads();

    const float* hb = H + (size_t)e * CAP * FF;
    const size_t wbase = (size_t)e * FF * HD;
    v8f acc[4] = {};

    const int nc    = n0w + (lane & 15);
    const int khalf = (lane >> 4) * 2;

    for (int k0 = 0; k0 < FF; k0 += 16) {
        __syncthreads();
        // cooperative A load: 64 rows x 16 cols
        #pragma unroll
        for (int i = 0; i < 4; i++) {
            const int m = (tid >> 4) + i * 16;
            const int k = tid & 15;
            float v = 0.f;
            if (tok[m] >= 0) v = hb[(size_t)(m0 + m) * FF + k0 + k];
            lds_a[m * 18 + k] = v;
        }
        __syncthreads();

        #pragma unroll
        for (int ks = 0; ks < 16; ks += 4) {
            const int kk = k0 + ks + khalf;
            const float* bp = Wd + wbase + (size_t)kk * HD + nc;
            v2f bf; bf.x = bp[0]; bf.y = bp[HD];
            #pragma unroll
            for (int ms = 0; ms < 4; ms++) {
                const float* ap =
                    &lds_a[(ms * 16 + (lane & 15)) * 18 + ks + khalf];
                v2f af; af.x = ap[0]; af.y = ap[1];
                acc[ms] = __builtin_amdgcn_wmma_f32_16x16x4_f32(
                    false, af, false, bf, (short)0, acc[ms], false, false);
            }
        }
    }

    const int colbase = n0w + (lane & 15);
    const int rowoff  = (lane >> 4) * 8;
    #pragma unroll
    for (int ms = 0; ms < 4; ms++) {
        #pragma unroll
        for (int j = 0; j < 8; j++) {
            const int row = m0 + ms * 16 + rowoff + j;
            if (row < ce) {
                const int t = tok[row - m0];
                const float s = scores[t];
                out[(size_t)t * HD + colbase] = acc[ms][j] * s;
            }
        }
    }
}

// ---------------------------------------------------------------------------
extern "C" void kernel_launch(void* const* d_in, const int* in_sizes, int n_in,
                              void* d_out, int out_size, void* d_ws,
                              size_t ws_size, hipStream_t stream) {
    const float* x           = (const float*)d_in[0];
    const int*   topk_idx    = (const int*)d_in[1];
    const float* topk_scores = (const float*)d_in[2];
    const float* Wg          = (const float*)d_in[3];
    const float* Wu          = (const float*)d_in[4];
    const float* Wd          = (const float*)d_in[5];
    float* out = (float*)d_out;

    // workspace layout: rows[NE*CAP] ints | cnt[NE] ints | pad | H floats
    char* ws = (char*)d_ws;
    int* rows = (int*)ws;
    int* cnt  = rows + NE * CAP;
    size_t hdr = ((size_t)(NE * CAP + NE) * sizeof(int) + 255) & ~(size_t)255;
    float* H = (float*)(ws + hdr);

    moe_zero_kernel<<<2048, 256, 0, stream>>>((float4*)out,
                                              (size_t)NT * HD / 4);
    moe_dispatch_kernel<<<NE, 256, 0, stream>>>(topk_idx, rows, cnt);

    dim3 g1(CAP / 64, (FF + 127) / 128, NE);   // (10, 6, 16)
    moe_gemm1_kernel<<<g1, 256, 0, stream>>>(x, Wg, Wu, rows, cnt, H);

    dim3 g2(CAP / 64, HD / 128, NE);           // (10, 32, 16)
    moe_gemm2_kernel<<<g2, 256, 0, stream>>>(H, Wd, rows, cnt, topk_scores,
                                             out);
}